// FS_Sampler_5892695130401
// MI455X (gfx1250) — compile-verified
//
#include <hip/hip_runtime.h>
#include <hip/hip_bf16.h>
#include <float.h>

// FPS sampler for MI455X (gfx1250).
// Heavy math = Gram matrix F*F^T (F = [8192 x 68] padded features) done with
// f32 WMMA 16x16x4; FPS itself is a sequential 2048-step chain done by one
// 1024-thread workgroup per batch with the running min-dist buffer in LDS.

typedef __attribute__((ext_vector_type(2))) float v2f;
typedef __attribute__((ext_vector_type(8))) float v8f;

#define NPTS   8192
#define NFEAT  64
#define KDIM   68          // 3 + 64, padded to 68 = 17*4 (K chunks of 4)
#define NBATCH 2
#define FPS_T  1024
#define PER    (NPTS / FPS_T)   // 8 points per thread

// ---------------------------------------------------------------------------
// Build padded feature rows feats[b][n][0..67] = {xyz, 64 feats, 0} and norms.
// ---------------------------------------------------------------------------
__global__ __launch_bounds__(256) void build_feats(const float* __restrict__ P,
                                                   const float* __restrict__ F,
                                                   float* __restrict__ feats,
                                                   float* __restrict__ S) {
  int gid = blockIdx.x * blockDim.x + threadIdx.x;
  if (gid >= NBATCH * NPTS) return;
  int b = gid / NPTS, n = gid % NPTS;
  const float* Pb = P + ((size_t)b * NPTS + n) * 3;
  const float* Fb = F + (size_t)b * NFEAT * NPTS + n;
  float* row = feats + (size_t)gid * KDIM;
  float acc = 0.f;
#pragma unroll
  for (int k = 0; k < 3; ++k) { float v = Pb[k]; row[k] = v; acc += v * v; }
  for (int c = 0; c < NFEAT; ++c) {
    float v = Fb[(size_t)c * NPTS];
    row[3 + c] = v;
    acc += v * v;
  }
  row[67] = 0.f;
  S[gid] = acc;
}

// ---------------------------------------------------------------------------
// G = F * F^T per batch via V_WMMA_F32_16X16X4_F32. One wave per 16x16 tile.
// A 16x4 f32 layout (ISA 7.12.2): lanes 0-15 -> M, VGPR0 holds K=0|2 across
// lane halves, VGPR1 holds K=1|3. B 4x16 mirrors with N across lanes.
// ---------------------------------------------------------------------------
__global__ __launch_bounds__(32) void gram_wmma(const float* __restrict__ feats,
                                                float* __restrict__ G) {
  int tileN = blockIdx.x;           // column tile
  int tileM = blockIdx.y;           // row tile
  int b     = blockIdx.z;
  const float* Fb = feats + (size_t)b * NPTS * KDIM;

  int lane = threadIdx.x;
  int half = lane >> 4;             // 0: K={0,1}, 1: K={2,3} within chunk
  int l    = lane & 15;

  const float* pa = Fb + (size_t)(tileM * 16 + l) * KDIM + 2 * half; // A: M = l
  const float* pb = Fb + (size_t)(tileN * 16 + l) * KDIM + 2 * half; // B: N = l

  v8f c = {};
#pragma unroll
  for (int kk = 0; kk < KDIM / 4; ++kk) {
    v2f a  = *(const v2f*)(pa + 4 * kk);   // {K0+2h, K0+2h+1}
    v2f bb = *(const v2f*)(pb + 4 * kk);
    c = __builtin_amdgcn_wmma_f32_16x16x4_f32(
        /*neg_a=*/false, a, /*neg_b=*/false, bb,
        /*c_mod=*/(short)0, c, /*reuse_a=*/false, /*reuse_b=*/false);
  }

  float* Gb = G + (size_t)b * NPTS * NPTS;
  int col = tileN * 16 + l;
#pragma unroll
  for (int r = 0; r < 8; ++r) {
    int row = tileM * 16 + r + 8 * half;   // C/D: VGPR r, lane half selects M|M+8
    Gb[(size_t)row * NPTS + col] = c[r];
  }
}

// ---------------------------------------------------------------------------
// Shared argmax helper pieces are inlined in each FPS kernel (tie-break: lowest
// index wins, matching jnp.argmax first-occurrence semantics).
// ---------------------------------------------------------------------------

// F-FPS using precomputed Gram rows: d(far,j) = S[far] + S[j] - 2*G[far][j]
__global__ __launch_bounds__(FPS_T) void fps_from_gram(const float* __restrict__ G,
                                                       const float* __restrict__ S,
                                                       int* __restrict__ out,
                                                       int npoint) {
  __shared__ float md[NPTS];
  __shared__ float rv[FPS_T];
  __shared__ int   ri[FPS_T];
  __shared__ int   s_far;

  int b = blockIdx.x;
  int t = threadIdx.x;
  const float* Gb = G + (size_t)b * NPTS * NPTS;
  const float* Sb = S + (size_t)b * NPTS;
  int* outb = out + (size_t)b * 2 * npoint;

  for (int j = t; j < NPTS; j += FPS_T) md[j] = 1e10f;
  if (t == 0) s_far = 0;
  __syncthreads();

  for (int s = 0; s < npoint; ++s) {
    int far = s_far;
    if (t == 0) outb[s] = far;
    float Sf = Sb[far];
    const float* row = Gb + (size_t)far * NPTS;

    float bestv = -FLT_MAX;
    int   besti = 0;
    int base = t * PER;
#pragma unroll
    for (int u = 0; u < PER; ++u) {
      int j = base + u;
      float d = Sf + Sb[j] - 2.0f * row[j];
      float m = fminf(md[j], d);
      md[j] = m;
      if (m > bestv) { bestv = m; besti = j; }
    }
    rv[t] = bestv; ri[t] = besti;
    __syncthreads();
    for (int off = FPS_T / 2; off > 0; off >>= 1) {
      if (t < off) {
        float v2 = rv[t + off]; int i2 = ri[t + off];
        if (v2 > rv[t] || (v2 == rv[t] && i2 < ri[t])) { rv[t] = v2; ri[t] = i2; }
      }
      __syncthreads();
    }
    if (t == 0) s_far = ri[0];
    __syncthreads();
  }
}

// F-FPS fallback (workspace too small for Gram): distances on the fly from the
// raw concat(points, features) representation; zero global scratch needed.
__global__ __launch_bounds__(FPS_T) void fps_feat_direct(const float* __restrict__ P,
                                                         const float* __restrict__ F,
                                                         int* __restrict__ out,
                                                         int npoint) {
  __shared__ float md[NPTS];
  __shared__ float rv[FPS_T];
  __shared__ int   ri[FPS_T];
  __shared__ float ffar[3 + NFEAT];
  __shared__ int   s_far;

  int b = blockIdx.x;
  int t = threadIdx.x;
  const float* Pb = P + (size_t)b * NPTS * 3;
  const float* Fb = F + (size_t)b * NFEAT * NPTS;
  int* outb = out + (size_t)b * 2 * npoint;

  for (int j = t; j < NPTS; j += FPS_T) md[j] = 1e10f;
  if (t == 0) s_far = 0;
  __syncthreads();

  for (int s = 0; s < npoint; ++s) {
    int far = s_far;
    if (t == 0) outb[s] = far;
    if (t < 3)            ffar[t] = Pb[(size_t)far * 3 + t];
    else if (t < 3 + NFEAT) ffar[t] = Fb[(size_t)(t - 3) * NPTS + far];
    __syncthreads();

    int base = t * PER;
    float d[PER];
#pragma unroll
    for (int u = 0; u < PER; ++u) {
      int j = base + u;
      float dx = Pb[(size_t)j * 3 + 0] - ffar[0];
      float dy = Pb[(size_t)j * 3 + 1] - ffar[1];
      float dz = Pb[(size_t)j * 3 + 2] - ffar[2];
      d[u] = dx * dx + dy * dy + dz * dz;
    }
    for (int c = 0; c < NFEAT; ++c) {
      float fc = ffar[3 + c];
      const float* Fc = Fb + (size_t)c * NPTS + base;
#pragma unroll
      for (int u = 0; u < PER; ++u) {
        float v = Fc[u] - fc;
        d[u] += v * v;
      }
    }

    float bestv = -FLT_MAX;
    int   besti = 0;
#pragma unroll
    for (int u = 0; u < PER; ++u) {
      int j = base + u;
      float m = fminf(md[j], d[u]);
      md[j] = m;
      if (m > bestv) { bestv = m; besti = j; }
    }
    rv[t] = bestv; ri[t] = besti;
    __syncthreads();
    for (int off = FPS_T / 2; off > 0; off >>= 1) {
      if (t < off) {
        float v2 = rv[t + off]; int i2 = ri[t + off];
        if (v2 > rv[t] || (v2 == rv[t] && i2 < ri[t])) { rv[t] = v2; ri[t] = i2; }
      }
      __syncthreads();
    }
    if (t == 0) s_far = ri[0];
    __syncthreads();
  }
}

// D-FPS on xyz only.
__global__ __launch_bounds__(FPS_T) void fps_xyz(const float* __restrict__ P,
                                                 int* __restrict__ out,
                                                 int npoint) {
  __shared__ float md[NPTS];
  __shared__ float rv[FPS_T];
  __shared__ int   ri[FPS_T];
  __shared__ int   s_far;

  int b = blockIdx.x;
  int t = threadIdx.x;
  const float* Pb = P + (size_t)b * NPTS * 3;
  int* outb = out + (size_t)b * 2 * npoint + npoint;   // second half: D-FPS

  for (int j = t; j < NPTS; j += FPS_T) md[j] = 1e10f;
  if (t == 0) s_far = 0;
  __syncthreads();

  for (int s = 0; s < npoint; ++s) {
    int far = s_far;
    if (t == 0) outb[s] = far;
    float xf = Pb[(size_t)far * 3 + 0];
    float yf = Pb[(size_t)far * 3 + 1];
    float zf = Pb[(size_t)far * 3 + 2];

    float bestv = -FLT_MAX;
    int   besti = 0;
    int base = t * PER;
#pragma unroll
    for (int u = 0; u < PER; ++u) {
      int j = base + u;
      float dx = Pb[(size_t)j * 3 + 0] - xf;
      float dy = Pb[(size_t)j * 3 + 1] - yf;
      float dz = Pb[(size_t)j * 3 + 2] - zf;
      float d = dx * dx + dy * dy + dz * dz;
      float m = fminf(md[j], d);
      md[j] = m;
      if (m > bestv) { bestv = m; besti = j; }
    }
    rv[t] = bestv; ri[t] = besti;
    __syncthreads();
    for (int off = FPS_T / 2; off > 0; off >>= 1) {
      if (t < off) {
        float v2 = rv[t + off]; int i2 = ri[t + off];
        if (v2 > rv[t] || (v2 == rv[t] && i2 < ri[t])) { rv[t] = v2; ri[t] = i2; }
      }
      __syncthreads();
    }
    if (t == 0) s_far = ri[0];
    __syncthreads();
  }
}

// ---------------------------------------------------------------------------
extern "C" void kernel_launch(void* const* d_in, const int* in_sizes, int n_in,
                              void* d_out, int out_size, void* d_ws, size_t ws_size,
                              hipStream_t stream) {
  const float* points   = (const float*)d_in[0];   // [2, 8192, 3]
  const float* features = (const float*)d_in[1];   // [2, 64, 8192]
  int* out = (int*)d_out;                          // [2, 2*npoint] int32 bits
  int npoint = out_size / (2 * NBATCH);            // 2048

  const size_t gBytes = (size_t)NBATCH * NPTS * NPTS * sizeof(float);   // 512 MB
  const size_t fBytes = (size_t)NBATCH * NPTS * KDIM * sizeof(float);
  const size_t sBytes = (size_t)NBATCH * NPTS * sizeof(float);

  if (ws_size >= gBytes + fBytes + sBytes) {
    // Fast path: WMMA Gram matrix, then row-gather FPS.
    float* G     = (float*)d_ws;
    float* feats = (float*)((char*)d_ws + gBytes);
    float* S     = (float*)((char*)d_ws + gBytes + fBytes);

    build_feats<<<(NBATCH * NPTS + 255) / 256, 256, 0, stream>>>(points, features, feats, S);
    dim3 grid(NPTS / 16, NPTS / 16, NBATCH);
    gram_wmma<<<grid, 32, 0, stream>>>(feats, G);
    fps_from_gram<<<NBATCH, FPS_T, 0, stream>>>(G, S, out, npoint);
  } else {
    // Fallback: on-the-fly feature distances, no global scratch.
    fps_feat_direct<<<NBATCH, FPS_T, 0, stream>>>(points, features, out, npoint);
  }
  fps_xyz<<<NBATCH, FPS_T, 0, stream>>>(points, out, npoint);
}